// WKPooling_47966194761766
// MI455X (gfx1250) — compile-verified
//
#include <hip/hip_runtime.h>
#include <hip/hip_bf16.h>
#include <math.h>

typedef __attribute__((ext_vector_type(2))) float v2f;
typedef __attribute__((ext_vector_type(4))) float v4f;
typedef __attribute__((ext_vector_type(8))) float v8f;

#define B_      8
#define S_      512
#define D_      768
#define L_      9
#define LAYER0  4
#define FSTRIDE 772      // padded LDS row stride (floats): 772%64=4 -> bank spread, 16B aligned
#define EPS_    1e-8f

// ---------------------------------------------------------------------------
// Kernel A: per-batch mask length, zero accumulators
// ---------------------------------------------------------------------------
__global__ __launch_bounds__(512) void wk_init(const int* __restrict__ mask,
                                               float* __restrict__ acc,
                                               float* __restrict__ wsum,
                                               int* __restrict__ lenm1)
{
    __shared__ int red[512];
    const int b = blockIdx.x;
    const int t = threadIdx.x;
    red[t] = mask[b * S_ + t];
    __syncthreads();
    for (int off = 256; off > 0; off >>= 1) {
        if (t < off) red[t] += red[t + off];
        __syncthreads();
    }
    if (t == 0) { lenm1[b] = red[0] - 1; wsum[b] = 0.0f; }
    acc[b * D_ + t] = 0.0f;
    if (t < (D_ - 512)) acc[b * D_ + 512 + t] = 0.0f;
}

// ---------------------------------------------------------------------------
// Kernel B: one block per token (b,s).
//  stage 1: global -> LDS fill of F (9 x 768 f32) + one zeroed pad row
//  stage 2: wave 0 computes G = F F^T via v_wmma_f32_16x16x4_f32
//           (192 K-chunks, 4 independent accumulators, unconditional loads)
//  stage 3: thread 0: per-k Cholesky of Gram submatrices -> align/nov -> alpha,
//           adjacent-cosine variance -> token weight w
//  stage 4: token_emb = alpha^T F from LDS; atomic accumulate w*token_emb
// ---------------------------------------------------------------------------
__global__ __launch_bounds__(256) void wk_token(const float* __restrict__ hs,
                                                const int* __restrict__ lenm1,
                                                float* __restrict__ acc,
                                                float* __restrict__ wsum)
{
    __shared__ float Fs[(L_ + 1) * FSTRIDE];   // row L_ is an all-zero pad row
    __shared__ float G16[16 * 16];
    __shared__ float alphaS[L_];
    __shared__ float wS;

    const int s   = blockIdx.x;
    const int b   = blockIdx.y;
    const int tid = threadIdx.x;

    // ---- stage 1: cooperative load, float4 granularity -------------------
    for (int idx = tid; idx < L_ * (D_ / 4); idx += 256) {
        const int l = idx / (D_ / 4);
        const int q = idx % (D_ / 4);
        const float* src = hs + (((size_t)(LAYER0 + l) * B_ + b) * S_ + s) * (size_t)D_ + q * 4;
        v4f v = *(const v4f*)src;
        *(v4f*)&Fs[l * FSTRIDE + q * 4] = v;
    }
    // zero pad row (read by lanes whose M index >= 9)
    for (int q = tid; q < D_ / 4; q += 256) {
        v4f z = {0.f, 0.f, 0.f, 0.f};
        *(v4f*)&Fs[L_ * FSTRIDE + q * 4] = z;
    }
    __syncthreads();

    // ---- stage 2: Gram via f32 WMMA (wave 0 only, EXEC all-ones) ---------
    if (tid < 32) {
        const int row  = tid & 15;                 // M index (A) / N index (D)
        const int kh   = tid >> 4;                 // K half of the 4-chunk
        const int rowe = (row < L_) ? row : L_;    // clamp to zero pad row
        const int base = rowe * FSTRIDE + kh * 2;  // per-lane LDS float offset

        v8f c0 = {0.f, 0.f, 0.f, 0.f, 0.f, 0.f, 0.f, 0.f};
        v8f c1 = c0, c2 = c0, c3 = c0;

        #pragma unroll 4
        for (int ch = 0; ch < D_ / 4; ch += 4) {
            v2f a0 = *(const v2f*)&Fs[base + (ch + 0) * 4];
            v2f a1 = *(const v2f*)&Fs[base + (ch + 1) * 4];
            v2f a2 = *(const v2f*)&Fs[base + (ch + 2) * 4];
            v2f a3 = *(const v2f*)&Fs[base + (ch + 3) * 4];
            // Gram: A-fragment and B-fragment register images are identical
            c0 = __builtin_amdgcn_wmma_f32_16x16x4_f32(false, a0, false, a0, (short)0, c0, false, false);
            c1 = __builtin_amdgcn_wmma_f32_16x16x4_f32(false, a1, false, a1, (short)0, c1, false, false);
            c2 = __builtin_amdgcn_wmma_f32_16x16x4_f32(false, a2, false, a2, (short)0, c2, false, false);
            c3 = __builtin_amdgcn_wmma_f32_16x16x4_f32(false, a3, false, a3, (short)0, c3, false, false);
        }
        v8f cacc = (c0 + c1) + (c2 + c3);

        #pragma unroll
        for (int r = 0; r < 8; ++r)
            G16[(r + kh * 8) * 16 + row] = cacc[r];   // D: M=r+8*kh, N=row
    }
    __syncthreads();

    // ---- stage 3: tiny per-token scalar math on thread 0 -----------------
    if (tid == 0) {
        float G[9][9];
        for (int i = 0; i < 9; ++i)
            for (int j = 0; j < 9; ++j)
                G[i][j] = G16[i * 16 + j];

        // adjacent-layer cosine variance (ddof=1)
        float c[8];
        float mean = 0.f;
        for (int l = 0; l < 8; ++l) {
            float den = sqrtf(G[l][l]) * sqrtf(G[l + 1][l + 1]);
            den = fmaxf(den, EPS_);
            c[l] = G[l][l + 1] / den;
            mean += c[l];
        }
        mean *= (1.0f / 8.0f);
        float var = 0.f;
        for (int l = 0; l < 8; ++l) { float d = c[l] - mean; var += d * d; }
        var *= (1.0f / 7.0f);

        // window index sets (left, right, k last) and their sizes
        const int ns[9] = {3, 3, 5, 5, 5, 5, 5, 4, 3};
        const int idxs[9][5] = {
            {1, 2, 0, 0, 0}, {2, 3, 1, 0, 0},
            {0, 1, 3, 4, 2}, {1, 2, 4, 5, 3}, {2, 3, 5, 6, 4},
            {3, 4, 6, 7, 5}, {4, 5, 7, 8, 6},
            {5, 6, 8, 7, 0}, {6, 7, 8, 0, 0}};

        float alignv[9], novv[9];
        for (int k = 0; k < 9; ++k) {
            const int n = ns[k];
            float M[5][5];
            for (int i = 0; i < n; ++i)
                for (int j = 0; j < n; ++j)
                    M[i][j] = G[idxs[k][i]][idxs[k][j]];

            // upper-triangular Cholesky: R^T R = M  (row-sign-equivalent to QR's R)
            float R[5][5];
            for (int i = 0; i < 5; ++i)
                for (int j = 0; j < 5; ++j) R[i][j] = 0.f;
            for (int i = 0; i < n; ++i) {
                float sdi = M[i][i];
                for (int t2 = 0; t2 < i; ++t2) sdi -= R[t2][i] * R[t2][i];
                float dii = sqrtf(fmaxf(sdi, 0.0f));
                R[i][i] = dii;
                float inv = 1.0f / fmaxf(dii, 1e-20f);
                for (int j = i + 1; j < n; ++j) {
                    float v = M[i][j];
                    for (int t2 = 0; t2 < i; ++t2) v -= R[t2][i] * R[t2][j];
                    R[i][j] = v * inv;
                }
            }

            float rn2 = 0.f, rs2 = 0.f;
            for (int i = 0; i < n; ++i)     rn2 += R[i][n - 1] * R[i][n - 1];
            for (int i = 0; i < n - 1; ++i) rs2 += R[i][n - 1] * R[i][n - 1];
            novv[k] = fabsf(R[n - 1][n - 1]) / fmaxf(sqrtf(rn2), 1e-20f);

            // column norms of Rsub
            float cn[4];
            for (int j = 0; j < n - 1; ++j) {
                float sum2 = 0.f;
                for (int i = 0; i <= j; ++i) sum2 += R[i][j] * R[i][j];
                cn[j] = fmaxf(sqrtf(sum2), 1e-20f);
            }
            float aacc = 0.f;
            const float invm = 1.0f / (float)(n - 1);
            for (int i = 0; i < n - 1; ++i) {
                float rm = 0.f;
                for (int j = 0; j < n - 1; ++j) rm += R[i][j] / cn[j];
                rm *= invm;
                aacc += rm * R[i][n - 1];
            }
            float al = aacc / fmaxf(sqrtf(rs2), 1e-20f);
            alignv[k] = 1.0f / (al * (float)(2 * n));
        }

        float sA = 0.f, sN = 0.f;
        for (int k = 0; k < 9; ++k) { sA += alignv[k]; sN += novv[k]; }
        float alpha[9], sAl = 0.f;
        for (int k = 0; k < 9; ++k) {
            alpha[k] = alignv[k] / sA + novv[k] / sN;
            sAl += alpha[k];
        }
        for (int k = 0; k < 9; ++k) alphaS[k] = alpha[k] / sAl;

        wS = (s < lenm1[b]) ? var : 0.0f;
    }
    __syncthreads();

    // ---- stage 4: weighted layer mix + batch accumulation ----------------
    const float w = wS;
    if (w != 0.0f) {
        for (int d = tid; d < D_; d += 256) {
            float te = 0.f;
            #pragma unroll
            for (int l = 0; l < L_; ++l) te += alphaS[l] * Fs[l * FSTRIDE + d];
            atomicAdd(&acc[b * D_ + d], w * te);
        }
        if (tid == 0) atomicAdd(&wsum[b], w);
    }
}

// ---------------------------------------------------------------------------
// Kernel C: out[b,d] = acc[b,d] / wsum[b]   (folds the v-normalization)
// ---------------------------------------------------------------------------
__global__ __launch_bounds__(768) void wk_finalize(const float* __restrict__ acc,
                                                   const float* __restrict__ wsum,
                                                   float* __restrict__ out)
{
    const int b = blockIdx.x;
    const int t = threadIdx.x;
    out[b * D_ + t] = acc[b * D_ + t] / wsum[b];
}

// ---------------------------------------------------------------------------
extern "C" void kernel_launch(void* const* d_in, const int* in_sizes, int n_in,
                              void* d_out, int out_size, void* d_ws, size_t ws_size,
                              hipStream_t stream)
{
    const float* hs   = (const float*)d_in[0];   // (13, 8, 512, 768) f32
    const int*   mask = (const int*)d_in[1];     // (8, 512) i32

    float* acc   = (float*)d_ws;                 // [B_*D_]
    float* wsum  = acc + B_ * D_;                // [B_]
    int*   lenm1 = (int*)(wsum + B_);            // [B_]

    wk_init<<<B_, 512, 0, stream>>>(mask, acc, wsum, lenm1);

    dim3 grid(S_, B_);
    wk_token<<<grid, 256, 0, stream>>>(hs, lenm1, acc, wsum);

    wk_finalize<<<B_, D_, 0, stream>>>(acc, wsum, (float*)d_out);
}